// Encoder_53102975647764
// MI455X (gfx1250) — compile-verified
//
#include <hip/hip_runtime.h>
#include <math.h>

// Problem constants (match reference)
#define H     512
#define T     1024
#define G4    4096   // 2 directions * 4H gate rows per layer
#define NBLK  64     // persistent blocks: each owns 16 hidden units (x2 dirs x32)
#define THR   256    // 64 gate rows * 4 K-slices

typedef float v2f __attribute__((ext_vector_type(2)));
typedef float v8f __attribute__((ext_vector_type(8)));

// ---------------------------------------------------------------------------
// Grid-wide sense-reversal barrier (all NBLK blocks co-resident).
// bar[0] = arrival counter, bar[1] = generation.
// ---------------------------------------------------------------------------
__device__ __forceinline__ void grid_sync(unsigned* bar, unsigned nblk) {
  __syncthreads();
  if (threadIdx.x == 0) {
    __threadfence();  // publish this block's stores at device scope
    unsigned gen = __hip_atomic_load(&bar[1], __ATOMIC_RELAXED, __HIP_MEMORY_SCOPE_AGENT);
    unsigned arr = __hip_atomic_fetch_add(&bar[0], 1u, __ATOMIC_ACQ_REL, __HIP_MEMORY_SCOPE_AGENT);
    if (arr == nblk - 1u) {
      __hip_atomic_store(&bar[0], 0u, __ATOMIC_RELAXED, __HIP_MEMORY_SCOPE_AGENT);
      __hip_atomic_fetch_add(&bar[1], 1u, __ATOMIC_RELEASE, __HIP_MEMORY_SCOPE_AGENT);
    } else {
      while (__hip_atomic_load(&bar[1], __ATOMIC_ACQUIRE, __HIP_MEMORY_SCOPE_AGENT) == gen) {
        __builtin_amdgcn_s_sleep(1);
      }
    }
  }
  __syncthreads();
}

__global__ void init_bar(unsigned* bar) {
  if (threadIdx.x < 2) bar[threadIdx.x] = 0u;
}

// ---------------------------------------------------------------------------
// Kernel 1: pre-gate GEMM with fp32 WMMA (V_WMMA_F32_16X16X4_F32).
//   pre0[t][g] = sum_k emb[x[t]][k] * w_ih0[g][k] + (b_ih0[g] + b_hh0[g])
// Block = 8 waves = 16 rows x 128 cols of output. The 16x512 A-tile (emb rows)
// is staged once into LDS (stride 516 floats: lanes m=0..15 hit banks 4m/4m+1
// for b64 reads -> conflict-free), then each wave runs K=512 in steps of 4.
// A (16x4 f32): lanes 0-15 M=0..15 K={k,k+1}; lanes 16-31 K={k+2,k+3}.
// B (4x16 f32): mirrored K split across half-waves, N = lane & 15.
// C/D (16x16 f32): VGPR r -> M=r (lanes 0-15) / M=r+8 (lanes 16-31), N=lane&15.
// ---------------------------------------------------------------------------
#define ASTRIDE 516

__global__ __launch_bounds__(256)
void pregate_wmma(const int* __restrict__ x,
                  const float* __restrict__ emb,
                  const float* __restrict__ w_ih0,
                  const float* __restrict__ b_ih0,
                  const float* __restrict__ b_hh0,
                  float* __restrict__ pre0) {
  __shared__ float shA[16 * ASTRIDE];
  __shared__ int   shx[16];

  const int t0 = blockIdx.y * 16;                 // timestep-row tile
  if (threadIdx.x < 16) shx[threadIdx.x] = x[t0 + threadIdx.x];
  __syncthreads();

  // Stage A-tile: 16 rows x 512 cols, float4 per thread-iteration.
  for (int idx = threadIdx.x; idx < 16 * 128; idx += 256) {
    int r  = idx >> 7;
    int c4 = (idx & 127) << 2;
    float4 v = *(const float4*)(emb + (size_t)shx[r] * H + c4);
    *(float4*)(shA + r * ASTRIDE + c4) = v;
  }
  __syncthreads();

  const int wave  = threadIdx.x >> 5;
  const int lane  = threadIdx.x & 31;             // wave32
  const int g0    = blockIdx.x * 128 + wave * 16; // gate-column tile
  const int m     = lane & 15;
  const int khalf = (lane >> 4) << 1;             // 0 or 2

  const float* arow = shA   + m * ASTRIDE + khalf;                // LDS
  const float* brow = w_ih0 + (size_t)(g0 + m) * H + khalf;       // L2

  v8f acc = {};
#pragma unroll 8
  for (int k = 0; k < H; k += 4) {
    v2f a = *(const v2f*)(arow + k);   // ds_load_b64
    v2f b = *(const v2f*)(brow + k);   // global_load_b64
    // 8 args: (neg_a, A, neg_b, B, c_mod, C, reuse_a, reuse_b)
    acc = __builtin_amdgcn_wmma_f32_16x16x4_f32(false, a, false, b,
                                                (short)0, acc, false, false);
  }

  const float bias = b_ih0[g0 + m] + b_hh0[g0 + m];
  const int   mb   = t0 + ((lane >> 4) << 3);     // M base: 0 or 8 within tile
#pragma unroll
  for (int r = 0; r < 8; ++r)
    pre0[(size_t)(mb + r) * G4 + g0 + m] = acc[r] + bias;
}

// ---------------------------------------------------------------------------
// Kernel 2: persistent recurrence, 2 grid-syncs per timestep.
// Block b owns direction d = b>>5 and hidden units j in [(b&31)*16, +16).
// Thread = (gate_local gl = tid>>2, K-slice ks = tid&3); gl = gate_type*16+jj.
// Gate row gr = d*2048 + gate_type*512 + j. All four gates of a hidden unit
// are block-local, so the cell update needs no grid barrier; c-state lives in
// LDS for the whole kernel. Only h0/h1 are published globally.
// ---------------------------------------------------------------------------
__device__ __forceinline__ float sigmoidf_(float v) {
  return 1.0f / (1.0f + __expf(-v));
}

// Contiguous fp32 weight slice (L2-resident) dotted against LDS vector.
// Four accumulators break the dependent-FMA chain.
__device__ __forceinline__ float dot_lds(const float* __restrict__ w,
                                         const float* s, int n) {
  float a0 = 0.f, a1 = 0.f, a2 = 0.f, a3 = 0.f;
#pragma unroll 4
  for (int k = 0; k < n; k += 4) {
    float4 wv = *(const float4*)(w + k);
    a0 = fmaf(wv.x, s[k],     a0);
    a1 = fmaf(wv.y, s[k + 1], a1);
    a2 = fmaf(wv.z, s[k + 2], a2);
    a3 = fmaf(wv.w, s[k + 3], a3);
  }
  return (a0 + a1) + (a2 + a3);
}

__global__ __launch_bounds__(THR)
void lstm_recurrence(const float* __restrict__ pre0,
                     const float* __restrict__ w_hh0,
                     const float* __restrict__ w_ih1,
                     const float* __restrict__ w_hh1,
                     const float* __restrict__ b_ih1,
                     const float* __restrict__ b_hh1,
                     float* __restrict__ out,
                     float* __restrict__ state,   // h0[1024] | h1[1024]
                     unsigned* bar) {
  float* h0 = state;          // [dir0 512 | dir1 512]
  float* h1 = state + 1024;

  __shared__ float sh[3 * H];    // staged h vectors (max concat(h0)+h1[d])
  __shared__ float red[THR];     // K-slice partials
  __shared__ float sg[64];       // reduced gate values for this block
  __shared__ float shc0[16];     // block-local cell state, layer 0
  __shared__ float shc1[16];     // block-local cell state, layer 1

  const int gl    = threadIdx.x >> 2;        // 0..63: gate_type*16 + jj
  const int ks    = threadIdx.x & 3;         // K-slice
  const int d     = blockIdx.x >> 5;         // direction
  const int jbase = (blockIdx.x & 31) * 16;  // hidden-unit base
  const int gt    = gl >> 4;                 // gate type (i,f,g,o)
  const int jj    = gl & 15;
  const int j     = jbase + jj;
  const int gr    = d * 2048 + gt * 512 + j; // global gate row

  // Zero-init carried state.
  if (threadIdx.x < 16) {
    shc0[threadIdx.x] = 0.f; shc1[threadIdx.x] = 0.f;
    h0[d * H + jbase + threadIdx.x] = 0.f;
    h1[d * H + jbase + threadIdx.x] = 0.f;
  }
  grid_sync(bar, NBLK);

  const float* wh0 = w_hh0 + (size_t)gr * H       + ks * 128;
  const float* wi1 = w_ih1 + (size_t)gr * (2 * H) + ks * 256;
  const float* wh1 = w_hh1 + (size_t)gr * H       + ks * 128;
  const float  bias1 = b_ih1[gr] + b_hh1[gr];

  for (int t = 0; t < T; ++t) {
    // ---- phase A: layer-0 gates (pre0 + W_hh0 @ h0[d]) + block-local cell --
    for (int i = threadIdx.x; i < H; i += THR) sh[i] = h0[d * H + i];
    __syncthreads();
    if (t + 1 < T && ks == 0)
      __builtin_prefetch(pre0 + (size_t)(t + 1) * G4 + gr, 0, 1);
    red[threadIdx.x] = dot_lds(wh0, sh + ks * 128, 128);
    __syncthreads();
    if (ks == 0) {
      int b4 = threadIdx.x;
      sg[gl] = pre0[(size_t)t * G4 + gr]
             + ((red[b4] + red[b4 + 1]) + (red[b4 + 2] + red[b4 + 3]));
    }
    __syncthreads();
    if (threadIdx.x < 16) {
      int u = threadIdx.x;
      float ig = sg[u], fg = sg[16 + u], gg = sg[32 + u], og = sg[48 + u];
      float c = sigmoidf_(fg) * shc0[u] + sigmoidf_(ig) * tanhf(gg);
      float h = sigmoidf_(og) * tanhf(c);
      shc0[u] = c;
      h0[d * H + jbase + u] = h;
    }
    grid_sync(bar, NBLK);   // publish h0

    // ---- phase B: layer-1 gates (W_ih1 @ concat(h0) + W_hh1 @ h1[d]) ------
    for (int i = threadIdx.x; i < 2 * H; i += THR) sh[i] = h0[i];
    for (int i = threadIdx.x; i < H; i += THR)     sh[2 * H + i] = h1[d * H + i];
    __syncthreads();
    red[threadIdx.x] = dot_lds(wi1, sh + ks * 256, 256)
                     + dot_lds(wh1, sh + 2 * H + ks * 128, 128);
    __syncthreads();
    if (ks == 0) {
      int b4 = threadIdx.x;
      sg[gl] = bias1 + ((red[b4] + red[b4 + 1]) + (red[b4 + 2] + red[b4 + 3]));
    }
    __syncthreads();
    if (threadIdx.x < 16) {
      int u = threadIdx.x;
      float ig = sg[u], fg = sg[16 + u], gg = sg[32 + u], og = sg[48 + u];
      float c = sigmoidf_(fg) * shc1[u] + sigmoidf_(ig) * tanhf(gg);
      float h = sigmoidf_(og) * tanhf(c);
      shc1[u] = c;
      h1[d * H + jbase + u] = h;
      out[(size_t)t * 1024 + d * H + jbase + u] = h;   // concat over dirs
    }
    grid_sync(bar, NBLK);   // publish h1
  }

  // Finals: hf[:L]+hf[L:] == [l0f+l1f, l0b+l1b] == h0+h1; same for c.
  if (threadIdx.x < 16) {
    int u  = threadIdx.x;
    int gi = d * H + jbase + u;
    out[(size_t)T * 1024 + gi]        = h0[gi] + h1[gi];
    out[(size_t)T * 1024 + 1024 + gi] = shc0[u] + shc1[u];
  }
}

// ---------------------------------------------------------------------------
// kernel_launch
// ---------------------------------------------------------------------------
extern "C" void kernel_launch(void* const* d_in, const int* in_sizes, int n_in,
                              void* d_out, int out_size, void* d_ws, size_t ws_size,
                              hipStream_t stream) {
  (void)in_sizes; (void)n_in; (void)out_size; (void)ws_size;
  const int*   x     = (const int*)  d_in[0];
  const float* emb   = (const float*)d_in[1];
  const float* w_ih0 = (const float*)d_in[2];
  const float* w_hh0 = (const float*)d_in[3];
  const float* b_ih0 = (const float*)d_in[4];
  const float* b_hh0 = (const float*)d_in[5];
  const float* w_ih1 = (const float*)d_in[6];
  const float* w_hh1 = (const float*)d_in[7];
  const float* b_ih1 = (const float*)d_in[8];
  const float* b_hh1 = (const float*)d_in[9];
  float* out = (float*)d_out;

  // ws layout: [0,64)B barrier | pre0: T*G4 f32 (16.8MB) | h-state: 2048 f32
  unsigned* bar   = (unsigned*)d_ws;
  float*    pre0  = (float*)d_ws + 16;
  float*    state = pre0 + (size_t)T * G4;

  hipLaunchKernelGGL(init_bar, dim3(1), dim3(32), 0, stream, bar);
  hipLaunchKernelGGL(pregate_wmma, dim3(G4 / 128, T / 16), dim3(256), 0, stream,
                     x, emb, w_ih0, b_ih0, b_hh0, pre0);
  hipLaunchKernelGGL(lstm_recurrence, dim3(NBLK), dim3(THR), 0, stream,
                     pre0, w_hh0, w_ih1, w_hh1, b_ih1, b_hh1, out, state, bar);
}